// LinearAttention_34179349741874
// MI455X (gfx1250) — compile-verified
//
#include <hip/hip_runtime.h>
#include <math.h>

// Problem constants (from reference)
#define BB 2
#define LL 4096
#define EE 1024
#define HH 16
#define DD 64          // head dim
#define NC 16          // chunks of 256 per sequence
#define HEADSZ ((size_t)BB * HH * LL * DD)   // 8,388,608 floats per Q/K/V buffer

typedef __attribute__((ext_vector_type(2))) float v2f;
typedef __attribute__((ext_vector_type(8))) float v8f;

// fp32 WMMA: D(16x16) = A(16x4) * B(4x16) + C   -- exact fp32 math on matrix cores
__device__ __forceinline__ v8f wmma4(v2f a, v2f b, v8f c) {
  return __builtin_amdgcn_wmma_f32_16x16x4_f32(false, a, false, b, (short)0, c,
                                               false, false);
}

// elu(x)+1 : x>0 ? x+1 : exp(x)
__device__ __forceinline__ float elup1(float x) {
  return x > 0.f ? x + 1.f : expf(x);
}

// ---------------------------------------------------------------------------
// Kernel 1: qkv = x @ W_qkv^T  (M=8192, N=3072, K=1024), fused activation +
// scatter into Q/K/V buffers laid out [B,H,L,D].
// Block = 256 threads (8 waves), computes a 128(M) x 64(N) output block.
// ---------------------------------------------------------------------------
__global__ __launch_bounds__(256) void qkv_gemm(const float* __restrict__ x,
                                                const float* __restrict__ W,
                                                float* __restrict__ qb,
                                                float* __restrict__ kb,
                                                float* __restrict__ vb) {
  __shared__ float xs[128 * 36];   // 128 x 32 slab, stride 36 (pad)
  __shared__ float wls[64 * 36];   // 64 x 32 slab

  const int tid = threadIdx.x;
  const int lane = tid & 31, wv = tid >> 5;
  const int hi = lane >> 4, nlo = lane & 15;
  const int mBase = blockIdx.y * 128;
  const int nBase = blockIdx.x * 64;

  v8f acc[4] = {};

  for (int kt = 0; kt < 1024; kt += 32) {
#pragma unroll
    for (int i = 0; i < 4; ++i) {
      int q = tid + 256 * i;
      int row = q >> 3, c4 = (q & 7) * 4;
      float4 t4 = *(const float4*)(x + (size_t)(mBase + row) * 1024 + kt + c4);
      *(float4*)(xs + row * 36 + c4) = t4;
    }
#pragma unroll
    for (int i = 0; i < 2; ++i) {
      int q = tid + 256 * i;
      int row = q >> 3, c4 = (q & 7) * 4;
      float4 t4 = *(const float4*)(W + (size_t)(nBase + row) * 1024 + kt + c4);
      *(float4*)(wls + row * 36 + c4) = t4;
    }
    __syncthreads();

    const float* xr = xs + (wv * 16 + nlo) * 36;
#pragma unroll
    for (int kk = 0; kk < 32; kk += 4) {
      v2f a;
      a.x = xr[kk + 2 * hi];
      a.y = xr[kk + 2 * hi + 1];
#pragma unroll
      for (int tc = 0; tc < 4; ++tc) {
        const float* wr = wls + (tc * 16 + nlo) * 36;
        v2f b;
        b.x = wr[kk + 2 * hi];
        b.y = wr[kk + 2 * hi + 1];
        acc[tc] = wmma4(a, b, acc[tc]);
      }
    }
    __syncthreads();
  }

  const int tsel = nBase >> 10;  // 0:q 1:k 2:v
  float* dst = (tsel == 0) ? qb : (tsel == 1) ? kb : vb;
#pragma unroll
  for (int tc = 0; tc < 4; ++tc) {
    int n = nBase + tc * 16 + nlo;
    int rem = n & 1023;
    int h = rem >> 6, d = rem & 63;
#pragma unroll
    for (int j = 0; j < 8; ++j) {
      int m = mBase + wv * 16 + j + 8 * hi;
      float y = acc[tc][j];
      if (tsel == 0)
        y = elup1(y) * 0.125f;   // scale = D^-0.5 = 0.125
      else if (tsel == 1)
        y = elup1(y);
      int b = m >> 12, l = m & 4095;
      dst[((size_t)(b * 16 + h) * 4096 + l) * 64 + d] = y;
    }
  }
}

#define AP 68  // padded row stride for 64-wide LDS tiles

// ---------------------------------------------------------------------------
// Kernel 2a: per-chunk stats. Block (c, bh) computes S_c = K_c^T V_c (64x64)
// and Z_c = colsum(K_c) for its 256-row chunk. Fully chunk-parallel (512 blocks).
// ---------------------------------------------------------------------------
__global__ __launch_bounds__(256) void chunk_stats(const float* __restrict__ kb,
                                                   const float* __restrict__ vb,
                                                   float* __restrict__ Sch,
                                                   float* __restrict__ Zch) {
  __shared__ float ks[64 * AP];
  __shared__ float vs[64 * AP];
  const int tid = threadIdx.x;
  const int lane = tid & 31, wv = tid >> 5;
  const int hi = lane >> 4, nlo = lane & 15;
  const int c = blockIdx.x, bh = blockIdx.y;
  const size_t base = ((size_t)bh * 4096 + c * 256) * 64;

  v8f sc[2] = {};
  float zacc = 0.f;

  for (int sb = 0; sb < 4; ++sb) {
#pragma unroll
    for (int i = 0; i < 4; ++i) {
      int q = tid + 256 * i;
      int row = q >> 4, c4 = (q & 15) * 4;
      size_t g = base + (size_t)(sb * 64 + row) * 64 + c4;
      *(float4*)(ks + row * AP + c4) = *(const float4*)(kb + g);
      *(float4*)(vs + row * AP + c4) = *(const float4*)(vb + g);
    }
    __syncthreads();
#pragma unroll
    for (int ti = 0; ti < 2; ++ti) {
      const int t = 2 * wv + ti, td = t >> 2, te = t & 3;
      v8f acc = sc[ti];
#pragma unroll
      for (int k0 = 0; k0 < 64; k0 += 4) {
        v2f a;  // A[d][m] = k[m][d]
        a.x = ks[(k0 + 2 * hi) * AP + td * 16 + nlo];
        a.y = ks[(k0 + 2 * hi + 1) * AP + td * 16 + nlo];
        v2f bv;
        bv.x = vs[(k0 + 2 * hi) * AP + te * 16 + nlo];
        bv.y = vs[(k0 + 2 * hi + 1) * AP + te * 16 + nlo];
        acc = wmma4(a, bv, acc);
      }
      sc[ti] = acc;
    }
    if (tid < 64) {
      float s = 0.f;
      for (int m = 0; m < 64; ++m) s += ks[m * AP + tid];
      zacc += s;
    }
    __syncthreads();
  }

  float* So = Sch + ((size_t)bh * NC + c) * 4096;
#pragma unroll
  for (int ti = 0; ti < 2; ++ti) {
    const int t = 2 * wv + ti, td = t >> 2, te = t & 3;
#pragma unroll
    for (int j = 0; j < 8; ++j)
      So[(td * 16 + j + 8 * hi) * 64 + te * 16 + nlo] = sc[ti][j];
  }
  if (tid < 64) Zch[((size_t)bh * NC + c) * 64 + tid] = zacc;
}

// ---------------------------------------------------------------------------
// Kernel 2b: in-place exclusive prefix over the 16 chunks of each head.
// Tiny (2.1M elements); one block per (b,h).
// ---------------------------------------------------------------------------
__global__ __launch_bounds__(256) void prefix_scan(float* __restrict__ Sch,
                                                   float* __restrict__ Zch) {
  const int bh = blockIdx.x, tid = threadIdx.x;
  for (int el = tid; el < 4096; el += 256) {
    float run = 0.f;
    for (int c = 0; c < NC; ++c) {
      size_t idx = ((size_t)bh * NC + c) * 4096 + el;
      float tmp = Sch[idx];
      Sch[idx] = run;
      run += tmp;
    }
  }
  if (tid < 64) {
    float run = 0.f;
    for (int c = 0; c < NC; ++c) {
      size_t idx = ((size_t)bh * NC + c) * 64 + tid;
      float tmp = Zch[idx];
      Zch[idx] = run;
      run += tmp;
    }
  }
}

// ---------------------------------------------------------------------------
// Kernel 2c: per-chunk attention. Block (c, bh) loads the exclusive-prefix
// state S,Z into LDS, then walks its four 64-row sub-chunks with the WMMA
// pipeline. Output overwrites the Q buffer. 512 independent blocks.
// ---------------------------------------------------------------------------
__global__ __launch_bounds__(256) void attn_chunk(float* __restrict__ qo,
                                                  const float* __restrict__ kb,
                                                  const float* __restrict__ vb,
                                                  const float* __restrict__ Sch,
                                                  const float* __restrict__ Zch) {
  extern __shared__ float sm[];
  float* qs = sm;                    // 64 x AP
  float* ks = sm + 64 * AP;
  float* vs = sm + 2 * 64 * AP;
  float* as_ = sm + 3 * 64 * AP;     // masked attn
  float* Ss = sm + 4 * 64 * AP;      // running state S
  float* Zs = sm + 5 * 64 * AP;      // running state Z (64)
  float* dens = Zs + 64;             // 64

  const int tid = threadIdx.x;
  const int lane = tid & 31, wv = tid >> 5;
  const int hi = lane >> 4, nlo = lane & 15;
  const int c = blockIdx.x, bh = blockIdx.y;
  const size_t base = (size_t)bh * 4096 * 64;

  // init S,Z from exclusive prefix
  const float* Sg = Sch + ((size_t)bh * NC + c) * 4096;
  for (int i = tid; i < 4096; i += 256) Ss[(i >> 6) * AP + (i & 63)] = Sg[i];
  if (tid < 64) Zs[tid] = Zch[((size_t)bh * NC + c) * 64 + tid];
  __syncthreads();

  for (int sb = 0; sb < 4; ++sb) {
    const int l0 = c * 256 + sb * 64;
    // ---- phase 1: stage q,k,v tiles (64x64 each) ----
#pragma unroll
    for (int i = 0; i < 4; ++i) {
      int q = tid + 256 * i;
      int row = q >> 4, c4 = (q & 15) * 4;
      size_t g = base + (size_t)(l0 + row) * 64 + c4;
      *(float4*)(qs + row * AP + c4) = *(const float4*)(qo + g);
      *(float4*)(ks + row * AP + c4) = *(const float4*)(kb + g);
      *(float4*)(vs + row * AP + c4) = *(const float4*)(vb + g);
    }
    __syncthreads();

    // ---- phase 2: attn = mask(q k^T) -> LDS ; num = q @ S (old S) ----
    v8f num[2];
#pragma unroll
    for (int ti = 0; ti < 2; ++ti) {
      const int t = 2 * wv + ti, tl = t >> 2, tc = t & 3;
      v8f acc = {};
      v8f nacc = {};
      const float* qr = qs + (tl * 16 + nlo) * AP;
      const float* kr = ks + (tc * 16 + nlo) * AP;
#pragma unroll
      for (int k0 = 0; k0 < 64; k0 += 4) {
        v2f a;
        a.x = qr[k0 + 2 * hi];
        a.y = qr[k0 + 2 * hi + 1];
        v2f bk;
        bk.x = kr[k0 + 2 * hi];
        bk.y = kr[k0 + 2 * hi + 1];
        acc = wmma4(a, bk, acc);
        v2f bs;
        bs.x = Ss[(k0 + 2 * hi) * AP + tc * 16 + nlo];
        bs.y = Ss[(k0 + 2 * hi + 1) * AP + tc * 16 + nlo];
        nacc = wmma4(a, bs, nacc);
      }
#pragma unroll
      for (int j = 0; j < 8; ++j) {
        int r = tl * 16 + j + 8 * hi;
        int cc = tc * 16 + nlo;
        as_[r * AP + cc] = (cc <= r) ? acc[j] : 0.f;  // causal mask
      }
      num[ti] = nacc;
    }
    __syncthreads();

    // ---- phase 3: den = rowsum(attn) + q@Z (old Z) ; num += attn @ v ----
    if (tid < 64) {
      float s = 0.f;
      const float* ar = as_ + tid * AP;
      for (int n = 0; n < 64; ++n) s += ar[n];
      const float* qr = qs + tid * AP;
      for (int d = 0; d < 64; ++d) s += qr[d] * Zs[d];
      dens[tid] = fmaxf(s, 1e-6f);
    }
#pragma unroll
    for (int ti = 0; ti < 2; ++ti) {
      const int t = 2 * wv + ti, tl = t >> 2, tc = t & 3;
      v8f nacc = num[ti];
      const float* ar = as_ + (tl * 16 + nlo) * AP;
#pragma unroll
      for (int k0 = 0; k0 < 64; k0 += 4) {
        v2f a;
        a.x = ar[k0 + 2 * hi];
        a.y = ar[k0 + 2 * hi + 1];
        v2f bv;
        bv.x = vs[(k0 + 2 * hi) * AP + tc * 16 + nlo];
        bv.y = vs[(k0 + 2 * hi + 1) * AP + tc * 16 + nlo];
        nacc = wmma4(a, bv, nacc);
      }
      num[ti] = nacc;
    }
    __syncthreads();

    // ---- phase 4: (if needed) S += k^T v ; Z += colsum(k) ; out = num/den ----
    if (sb != 3) {  // uniform branch; last sub-chunk's state is unused
#pragma unroll
      for (int ti = 0; ti < 2; ++ti) {
        const int t = 2 * wv + ti, td = t >> 2, te = t & 3;
        v8f scv;
#pragma unroll
        for (int j = 0; j < 8; ++j)
          scv[j] = Ss[(td * 16 + j + 8 * hi) * AP + te * 16 + nlo];
#pragma unroll
        for (int k0 = 0; k0 < 64; k0 += 4) {
          v2f a;  // A[d][m] = k[m][d]
          a.x = ks[(k0 + 2 * hi) * AP + td * 16 + nlo];
          a.y = ks[(k0 + 2 * hi + 1) * AP + td * 16 + nlo];
          v2f bv;
          bv.x = vs[(k0 + 2 * hi) * AP + te * 16 + nlo];
          bv.y = vs[(k0 + 2 * hi + 1) * AP + te * 16 + nlo];
          scv = wmma4(a, bv, scv);
        }
#pragma unroll
        for (int j = 0; j < 8; ++j)
          Ss[(td * 16 + j + 8 * hi) * AP + te * 16 + nlo] = scv[j];
      }
      if (tid < 64) {
        float s = Zs[tid];
        for (int m = 0; m < 64; ++m) s += ks[m * AP + tid];
        Zs[tid] = s;
      }
    }
#pragma unroll
    for (int ti = 0; ti < 2; ++ti) {
      const int t = 2 * wv + ti, tl = t >> 2, te = t & 3;
#pragma unroll
      for (int j = 0; j < 8; ++j) {
        int r = tl * 16 + j + 8 * hi;
        float v = num[ti][j] / dens[r];
        qo[base + (size_t)(l0 + r) * 64 + te * 16 + nlo] = v;  // O over Q
      }
    }
    __syncthreads();
  }
}

// ---------------------------------------------------------------------------
// Kernel 3: out = O @ W_out^T  (M=8192, N=1024, K=1024), O read from the
// head-interleaved [B,H,L,D] buffer (k = h*64+d).
// ---------------------------------------------------------------------------
__global__ __launch_bounds__(256) void out_gemm(const float* __restrict__ oh,
                                                const float* __restrict__ W,
                                                float* __restrict__ out) {
  __shared__ float xs[128 * 36];
  __shared__ float wls[64 * 36];

  const int tid = threadIdx.x;
  const int lane = tid & 31, wv = tid >> 5;
  const int hi = lane >> 4, nlo = lane & 15;
  const int mBase = blockIdx.y * 128;
  const int nBase = blockIdx.x * 64;

  v8f acc[4] = {};

  for (int kt = 0; kt < 1024; kt += 32) {
    const int h = kt >> 6;         // 32-wide slab never crosses a head
    const int dbase = kt & 63;
#pragma unroll
    for (int i = 0; i < 4; ++i) {
      int q = tid + 256 * i;
      int row = q >> 3, c4 = (q & 7) * 4;
      int m = mBase + row;
      int b = m >> 12, l = m & 4095;
      float4 t4 = *(const float4*)(oh + ((size_t)(b * 16 + h) * 4096 + l) * 64 +
                                   dbase + c4);
      *(float4*)(xs + row * 36 + c4) = t4;
    }
#pragma unroll
    for (int i = 0; i < 2; ++i) {
      int q = tid + 256 * i;
      int row = q >> 3, c4 = (q & 7) * 4;
      float4 t4 = *(const float4*)(W + (size_t)(nBase + row) * 1024 + kt + c4);
      *(float4*)(wls + row * 36 + c4) = t4;
    }
    __syncthreads();

    const float* xr = xs + (wv * 16 + nlo) * 36;
#pragma unroll
    for (int kk = 0; kk < 32; kk += 4) {
      v2f a;
      a.x = xr[kk + 2 * hi];
      a.y = xr[kk + 2 * hi + 1];
#pragma unroll
      for (int tc = 0; tc < 4; ++tc) {
        const float* wr = wls + (tc * 16 + nlo) * 36;
        v2f b;
        b.x = wr[kk + 2 * hi];
        b.y = wr[kk + 2 * hi + 1];
        acc[tc] = wmma4(a, b, acc[tc]);
      }
    }
    __syncthreads();
  }

#pragma unroll
  for (int tc = 0; tc < 4; ++tc) {
    int n = nBase + tc * 16 + nlo;
#pragma unroll
    for (int j = 0; j < 8; ++j) {
      int m = mBase + wv * 16 + j + 8 * hi;
      out[(size_t)m * 1024 + n] = acc[tc][j];
    }
  }
}

// ---------------------------------------------------------------------------
extern "C" void kernel_launch(void* const* d_in, const int* in_sizes, int n_in,
                              void* d_out, int out_size, void* d_ws,
                              size_t ws_size, hipStream_t stream) {
  const float* x = (const float*)d_in[0];      // [B,L,E]
  const float* Wqkv = (const float*)d_in[1];   // [3E,E]
  const float* Wout = (const float*)d_in[2];   // [E,E]
  float* out = (float*)d_out;                  // [B,L,E]

  float* qbuf = (float*)d_ws;                  // [B,H,L,D]; later holds O
  float* kbuf = qbuf + HEADSZ;
  float* vbuf = kbuf + HEADSZ;
  float* Sch = vbuf + HEADSZ;                  // [B*H, NC, 64,64]
  float* Zch = Sch + (size_t)BB * HH * NC * 4096;  // [B*H, NC, 64]

  // 1) QKV projection + feature map, scatter to head layout
  qkv_gemm<<<dim3(48, 64), 256, 0, stream>>>(x, Wqkv, qbuf, kbuf, vbuf);

  // 2) chunk-parallel linear attention
  chunk_stats<<<dim3(NC, BB * HH), 256, 0, stream>>>(kbuf, vbuf, Sch, Zch);
  prefix_scan<<<dim3(BB * HH), 256, 0, stream>>>(Sch, Zch);
  const size_t smem = (size_t)(5 * 64 * AP + 128) * sizeof(float);  // ~87.5 KB
  attn_chunk<<<dim3(NC, BB * HH), 256, smem, stream>>>(qbuf, kbuf, vbuf, Sch,
                                                       Zch);

  // 3) output projection
  out_gemm<<<dim3(16, 64), 256, 0, stream>>>(qbuf, Wout, out);
}